// HyperPatchNoPadding_32375463477394
// MI455X (gfx1250) — compile-verified
//
#include <hip/hip_runtime.h>
#include <hip/hip_bf16.h>

// Shapes (fixed by the reference):
//   x : [8, 64, 256, 256] fp32
//   s : [8, 4096, 16, 16] fp32   (hyper weights, 4096 = out*in = 64*64)
//   out:[8, 64, 256, 256] fp32
// Per patch (b, fi, fj): D[o, n] = sum_c W[o,c] * X[c, n],
//   W[o,c] = s[b, o*64+c, fi, fj]
//   X[c, p*16+q] = x[b, c, fi*16+p, fj*16+q]
//   out[b, o, fi*16+p, fj*16+q] = D[o, p*16+q]

typedef __attribute__((ext_vector_type(2))) float v2f;
typedef __attribute__((ext_vector_type(8))) float v8f;

__global__ __launch_bounds__(128)
void hyperpatch_wmma_f32(const float* __restrict__ x,
                         const float* __restrict__ s,
                         float* __restrict__ out)
{
    const int lane = threadIdx.x & 31;
    const int wave = threadIdx.x >> 5;   // 0..3 : output-channel tile
    const int patch = blockIdx.x;        // 0..2047
    const int b  = patch >> 8;           // /256
    const int fi = (patch >> 4) & 15;
    const int fj = patch & 15;

    const int l    = lane & 15;          // position within half-wave
    const int half = lane >> 4;          // 0: K%4 in {0,1}, 1: K%4 in {2,3}
    const int m0   = wave << 4;          // 16-row M tile

    // ---------------- A fragments: W_patch[m0+l, c] -------------------
    // s linear index: ((b*4096 + o*64 + c) * 256) + fi*16 + fj
    // A 16x4 fp32 layout (ISA 7.12.2): lanes 0-15 -> M=lane, VGPR0=K0,VGPR1=K1
    //                                  lanes16-31 -> M=lane-16, VGPR0=K2,VGPR1=K3
    const int o = m0 + l;
    const long sbase = ((long)b * 4096 + (long)o * 64) * 256 + fi * 16 + fj;
    v2f a[16];
#pragma unroll
    for (int kk = 0; kk < 16; ++kk) {
        const int c0 = kk * 4 + 2 * half;
        a[kk].x = s[sbase + (long)c0 * 256];
        a[kk].y = s[sbase + (long)(c0 + 1) * 256];
    }

    // x linear index: (b*64 + c)*65536 + h*256 + w ; h = fi*16+p, w = fj*16+q
    const long xbase = (long)b * 64 * 65536 + (long)(fi * 16) * 256 + fj * 16 + l;
    // out linear index base for this wave's M tile
    const long obase = ((long)b * 64 + m0 + half * 8) * 65536
                     + (long)(fi * 16) * 256 + fj * 16 + l;

#pragma unroll 1
    for (int p = 0; p < 16; ++p) {       // N tile = pixel row p (16 pixels)
        // -------- B fragments: X[c, p*16 + l], coalesced 64B per half-wave
        // B 4x16 fp32 layout mirrors A: lanes 0-15 rows K0/K1, lanes16-31 K2/K3
        const long xrow = xbase + (long)p * 256;
        v2f bf[16];
#pragma unroll
        for (int kk = 0; kk < 16; ++kk) {
            const int c0 = kk * 4 + 2 * half;
            bf[kk].x = x[xrow + (long)c0 * 65536];
            bf[kk].y = x[xrow + (long)(c0 + 1) * 65536];
        }

        // -------- accumulate K = 64 with 16 WMMA f32 16x16x4 ops
        v8f acc = {};
#pragma unroll
        for (int kk = 0; kk < 16; ++kk) {
            acc = __builtin_amdgcn_wmma_f32_16x16x4_f32(
                /*neg_a=*/false, a[kk],
                /*neg_b=*/false, bf[kk],
                /*c_mod=*/(short)0, acc,
                /*reuse_a=*/false, /*reuse_b=*/false);
        }

        // -------- store D tile: VGPR r -> M = r (+8 for upper half-wave)
        const long orow = obase + (long)p * 256;
#pragma unroll
        for (int r = 0; r < 8; ++r) {
            out[orow + (long)r * 65536] = acc[r];
        }
    }
}

extern "C" void kernel_launch(void* const* d_in, const int* in_sizes, int n_in,
                              void* d_out, int out_size, void* d_ws, size_t ws_size,
                              hipStream_t stream) {
    const float* x = (const float*)d_in[0];   // 8*64*256*256 floats
    const float* s = (const float*)d_in[1];   // 8*4096*16*16 floats
    float* out = (float*)d_out;               // 8*64*256*256 floats

    const int num_patches = 8 * 16 * 16;      // 2048 workgroups, 1 per patch
    hyperpatch_wmma_f32<<<num_patches, 128, 0, stream>>>(x, s, out);
}